// WaveletLayer_84576495993314
// MI455X (gfx1250) — compile-verified
//
#include <hip/hip_runtime.h>

// ---------------------------------------------------------------------------
// WaveletLayer on gfx1250 (MI455X):
//   deg -> edge weights -> 3x SpMM (float atomics, L2-resident) ->
//   4x accumulating fp32 WMMA GEMM passes (V_WMMA_F32_16X16X4_F32)
// ---------------------------------------------------------------------------

typedef __attribute__((ext_vector_type(2))) float v2f;
typedef __attribute__((ext_vector_type(8))) float v8f;

#define NN   100000
#define EE   600000
#define CIN  128
#define COUT 128
#define KHOP 3

// ---------------------------- utility kernels ------------------------------

__global__ void zero_f32(float* __restrict__ p, int n) {
    int i = blockIdx.x * blockDim.x + threadIdx.x;
    if (i < n) p[i] = 0.0f;
}

__global__ void zero_f32x4(float4* __restrict__ p, int n4) {
    int i = blockIdx.x * blockDim.x + threadIdx.x;
    if (i < n4) p[i] = make_float4(0.f, 0.f, 0.f, 0.f);
}

// degree counts: scatter-add 1.0 per edge endpoint (exact for counts < 2^24)
__global__ void degree_kernel(const long long* __restrict__ src,
                              const long long* __restrict__ dst,
                              float* __restrict__ deg_src,
                              float* __restrict__ deg_dst, int E) {
    int e = blockIdx.x * blockDim.x + threadIdx.x;
    if (e >= E) return;
    unsafeAtomicAdd(deg_src + (int)src[e], 1.0f);
    unsafeAtomicAdd(deg_dst + (int)dst[e], 1.0f);
}

// coef[k] = softmax(hop_logits)[k] * exp(-exp(log_scale) * k)   (k=0 -> 1.0 decay)
__global__ void coef_kernel(const float* __restrict__ log_scale,
                            const float* __restrict__ hop_logits,
                            float* __restrict__ coef) {
    if (threadIdx.x != 0 || blockIdx.x != 0) return;
    float s = __expf(log_scale[0]);
    float m = hop_logits[0];
    for (int k = 1; k <= KHOP; ++k) m = fmaxf(m, hop_logits[k]);
    float ex[KHOP + 1];
    float sum = 0.0f;
    for (int k = 0; k <= KHOP; ++k) { ex[k] = __expf(hop_logits[k] - m); sum += ex[k]; }
    float inv = 1.0f / sum;
    for (int k = 0; k <= KHOP; ++k) {
        float alpha = ex[k] * inv;
        float decay = (k == 0) ? 1.0f : __expf(-s * (float)k);
        coef[k] = alpha * decay;
    }
}

// ew[e] = dsi[src]*ddi[dst] with d^{-1/2}, zero for isolated endpoints
__global__ void edge_weight_kernel(const long long* __restrict__ src,
                                   const long long* __restrict__ dst,
                                   const float* __restrict__ deg_src,
                                   const float* __restrict__ deg_dst,
                                   float* __restrict__ ew, int E) {
    int e = blockIdx.x * blockDim.x + threadIdx.x;
    if (e >= E) return;
    float ds = deg_src[(int)src[e]];
    float dd = deg_dst[(int)dst[e]];
    float dsi = (ds > 0.0f) ? rsqrtf(fmaxf(ds, 1.0f)) : 0.0f;
    float ddi = (dd > 0.0f) ? rsqrtf(fmaxf(dd, 1.0f)) : 0.0f;
    ew[e] = dsi * ddi;
}

// -------------------------------- SpMM -------------------------------------
// next[dst] += ew[e] * cur[src]; one wave per edge, one float4 (4 ch) per lane.
__global__ void spmm_kernel(const float4* __restrict__ cur,
                            float4* __restrict__ next,
                            const long long* __restrict__ src,
                            const long long* __restrict__ dst,
                            const float* __restrict__ ew, int E) {
    int t = blockIdx.x * blockDim.x + threadIdx.x;
    int e = t >> 5;
    if (e >= E) return;
    int lane = t & 31;                      // 32 lanes * float4 = 128 channels
    int s = (int)src[e];
    int d = (int)dst[e];
    float w = ew[e];
    float4 v = cur[(size_t)s * 32 + lane];
    float* p = (float*)(next + (size_t)d * 32 + lane);
    unsafeAtomicAdd(p + 0, w * v.x);
    unsafeAtomicAdd(p + 1, w * v.y);
    unsafeAtomicAdd(p + 2, w * v.z);
    unsafeAtomicAdd(p + 3, w * v.w);
}

// ------------------------------ WMMA GEMM ----------------------------------
// out[16x16 tile] (+)= coef * A[16 x 128] @ W[128 x 16], fp32 WMMA, K-steps of 4.
// A-frag: lane L holds row M=L%16, K = 2*(L/16)+{0,1}  (v2f)
// B-frag: lane L holds col N=L%16, K = 2*(L/16)+{0,1}  (v2f)
// C/D:    lane L holds col N=L%16, rows M = j + 8*(L/16), j=0..7 (v8f)
__global__ void gemm_wmma_kernel(const float* __restrict__ A,
                                 const float* __restrict__ W,
                                 const float* __restrict__ bias,
                                 float* __restrict__ out,
                                 const float* __restrict__ coefArr,
                                 int hop, int initFlag) {
    const int lane  = threadIdx.x & 31;
    const int gwave = blockIdx.x * (blockDim.x >> 5) + (threadIdx.x >> 5);
    const int tileN = gwave & 7;            // 8 tiles along COUT=128
    const int tileM = gwave >> 3;           // 6250 tiles along N=100000
    const float coef = coefArr[hop];

    const int mbase = tileM * 16;
    const int nbase = tileN * 16;
    const int l15   = lane & 15;
    const int hi    = lane >> 4;            // 0 or 1
    const int khi   = hi * 2;               // K sub-offset per lane half
    const int ncol  = nbase + l15;
    const int rowOff = hi * 8;

    v8f c;
    if (initFlag) {
        const float bv = bias[ncol];
#pragma unroll
        for (int j = 0; j < 8; ++j) c[j] = bv;
    } else {
#pragma unroll
        for (int j = 0; j < 8; ++j)
            c[j] = out[(size_t)(mbase + rowOff + j) * COUT + ncol];
    }

    const float* arow = A + (size_t)(mbase + l15) * CIN + khi;
    const float* bcol = W + (size_t)khi * COUT + ncol;

#pragma unroll 4
    for (int k0 = 0; k0 < CIN; k0 += 4) {
        v2f av;
        av.x = coef * arow[k0];
        av.y = coef * arow[k0 + 1];
        v2f bv;
        bv.x = bcol[(size_t)k0 * COUT];
        bv.y = bcol[(size_t)(k0 + 1) * COUT];
        c = __builtin_amdgcn_wmma_f32_16x16x4_f32(
                /*neg_a=*/false, av, /*neg_b=*/false, bv,
                /*c_mod=*/(short)0, c, /*reuse_a=*/false, /*reuse_b=*/false);
    }

#pragma unroll
    for (int j = 0; j < 8; ++j)
        out[(size_t)(mbase + rowOff + j) * COUT + ncol] = c[j];
}

// ------------------------------- launcher ----------------------------------

extern "C" void kernel_launch(void* const* d_in, const int* in_sizes, int n_in,
                              void* d_out, int out_size, void* d_ws, size_t ws_size,
                              hipStream_t stream) {
    const float*     x          = (const float*)d_in[0];
    const long long* edge_index = (const long long*)d_in[1];
    // d_in[2] = num_nodes (unused; N is a compile-time constant)
    const float*     W          = (const float*)d_in[3];
    const float*     bias       = (const float*)d_in[4];
    const float*     log_scale  = (const float*)d_in[5];
    const float*     hop_logits = (const float*)d_in[6];
    float*           out        = (float*)d_out;
    float*           ws         = (float*)d_ws;

    const long long* src = edge_index;        // edge_index[0, :]
    const long long* dst = edge_index + EE;   // edge_index[1, :]

    // workspace layout (floats):
    float* deg_src = ws;                              // [N]
    float* deg_dst = ws + NN;                         // [N]
    float* ew      = ws + 2 * NN;                     // [E]
    float* coef    = ws + 2 * NN + EE;                // [KHOP+1]
    float* bufA    = ws + 2 * NN + EE + 4;            // [N*128] (16B aligned)
    float* bufB    = bufA + (size_t)NN * CIN;         // [N*128]

    const int TB = 256;

    // 1) degrees
    zero_f32<<<(2 * NN + TB - 1) / TB, TB, 0, stream>>>(deg_src, 2 * NN);
    degree_kernel<<<(EE + TB - 1) / TB, TB, 0, stream>>>(src, dst, deg_src, deg_dst, EE);

    // 2) hop coefficients + edge weights
    coef_kernel<<<1, 32, 0, stream>>>(log_scale, hop_logits, coef);
    edge_weight_kernel<<<(EE + TB - 1) / TB, TB, 0, stream>>>(src, dst, deg_src, deg_dst, ew, EE);

    // GEMM grid: 50000 waves (6250 M-tiles x 8 N-tiles), 8 waves per block
    const int gemmBlocks = (NN / 16) * (COUT / 16) / (TB / 32);   // 6250
    const int spmmBlocks = (EE * 32 + TB - 1) / TB;               // wave per edge
    const int zero4Blocks = (NN * (CIN / 4) + TB - 1) / TB;

    // 3) hop 0: out = bias + coef0 * x @ W
    gemm_wmma_kernel<<<gemmBlocks, TB, 0, stream>>>(x, W, bias, out, coef, 0, 1);

    // 4) hop 1: bufA = A_hat @ x ; out += coef1 * bufA @ W
    zero_f32x4<<<zero4Blocks, TB, 0, stream>>>((float4*)bufA, NN * (CIN / 4));
    spmm_kernel<<<spmmBlocks, TB, 0, stream>>>((const float4*)x, (float4*)bufA, src, dst, ew, EE);
    gemm_wmma_kernel<<<gemmBlocks, TB, 0, stream>>>(bufA, W, bias, out, coef, 1, 0);

    // 5) hop 2: bufB = A_hat @ bufA ; out += coef2 * bufB @ W
    zero_f32x4<<<zero4Blocks, TB, 0, stream>>>((float4*)bufB, NN * (CIN / 4));
    spmm_kernel<<<spmmBlocks, TB, 0, stream>>>((const float4*)bufA, (float4*)bufB, src, dst, ew, EE);
    gemm_wmma_kernel<<<gemmBlocks, TB, 0, stream>>>(bufB, W, bias, out, coef, 2, 0);

    // 6) hop 3: bufA = A_hat @ bufB ; out += coef3 * bufA @ W
    zero_f32x4<<<zero4Blocks, TB, 0, stream>>>((float4*)bufA, NN * (CIN / 4));
    spmm_kernel<<<spmmBlocks, TB, 0, stream>>>((const float4*)bufB, (float4*)bufA, src, dst, ew, EE);
    gemm_wmma_kernel<<<gemmBlocks, TB, 0, stream>>>(bufA, W, bias, out, coef, 3, 0);
}